// GATBlock_7559142441081
// MI455X (gfx1250) — compile-verified
//
#include <hip/hip_runtime.h>
#include <hip/hip_bf16.h>

typedef __attribute__((ext_vector_type(2))) float v2f;
typedef __attribute__((ext_vector_type(8))) float v8f;

#define IN_CH 64
#define OUT_DIM 128
#define HEADS 4

static __device__ __forceinline__ float lrelu(float x) { return fmaxf(x, 0.2f * x); }

// monotonic float<->uint key so unsigned atomicMax == float max
static __device__ __forceinline__ unsigned fkey(float f) {
  unsigned u = __float_as_uint(f);
  return (u & 0x80000000u) ? ~u : (u | 0x80000000u);
}
static __device__ __forceinline__ float fkey_inv(unsigned k) {
  return __uint_as_float((k & 0x80000000u) ? (k & 0x7FFFFFFFu) : ~k);
}

// ---------------------------------------------------------------------------
// K1: dual GEMM via V_WMMA_F32_16X16X4_F32.
//   xt = x @ lin_w^T   (cols 0..127 -> xt buffer)
//   z  = x @ res_w^T   (cols 0..127 -> z accumulator, biases added later)
// One wave = one 16x16 tile; 8 waves/block, 2 tiles/wave -> 256 cols per
// 16-node chunk. A-operand layout per ISA: lanes 0-15 rows M, kb=(lane>>4)*2.
// ---------------------------------------------------------------------------
__global__ __launch_bounds__(256) void k_gemm(const float* __restrict__ x,
                                              const float* __restrict__ lin_w,
                                              const float* __restrict__ res_w,
                                              float* __restrict__ xt,
                                              float* __restrict__ z,
                                              int N) {
  const int wave = threadIdx.x >> 5;
  const int lane = threadIdx.x & 31;
  const int m0 = blockIdx.x * 16;
  const int mrow = lane & 15;
  const int kb = (lane >> 4) << 1;  // 0 or 2
  int arow = m0 + mrow;
  if (arow >= N) arow = N - 1;  // clamp loads; stores are guarded
  const float* xr = x + (size_t)arow * IN_CH;
  v2f a[16];
#pragma unroll
  for (int ks = 0; ks < 16; ++ks) {
    a[ks].x = xr[ks * 4 + kb];
    a[ks].y = xr[ks * 4 + kb + 1];
  }
#pragma unroll
  for (int tt = 0; tt < 2; ++tt) {
    const int tile = wave + tt * 8;  // 0..15
    const float* W = (tile < 8) ? lin_w : res_w;
    const int cbase = (tile & 7) * 16;
    const float* wr = W + (size_t)(cbase + (lane & 15)) * IN_CH;  // W[col, :]
    v8f c = {0.f, 0.f, 0.f, 0.f, 0.f, 0.f, 0.f, 0.f};
#pragma unroll
    for (int ks = 0; ks < 16; ++ks) {
      v2f b;
      b.x = wr[ks * 4 + kb];
      b.y = wr[ks * 4 + kb + 1];
      c = __builtin_amdgcn_wmma_f32_16x16x4_f32(false, a[ks], false, b,
                                                (short)0, c, false, false);
    }
    float* dst = (tile < 8) ? xt : z;
    const int mb = m0 + ((lane >> 4) << 3);
#pragma unroll
    for (int r = 0; r < 8; ++r) {
      const int m = mb + r;
      if (m < N) dst[(size_t)m * OUT_DIM + cbase + (lane & 15)] = c[r];
    }
  }
}

// ---------------------------------------------------------------------------
// K2: per-node prep (one wave per node, 4 channels/lane):
//   a_src[n,h] = <xt[n,h,:], att_src[h,:]>, same for a_dst
//   z[n,:] += gat_bias + res_b   (z already holds x@res_w^T)
//   reset emax keys / esum (fresh every call -> replay safe)
// ---------------------------------------------------------------------------
__global__ __launch_bounds__(256) void k_node(const float* __restrict__ xt,
                                              const float* __restrict__ att_src,
                                              const float* __restrict__ att_dst,
                                              const float* __restrict__ gat_bias,
                                              const float* __restrict__ res_b,
                                              float* __restrict__ z,
                                              float* __restrict__ a_src,
                                              float* __restrict__ a_dst,
                                              unsigned* __restrict__ emax_u,
                                              float* __restrict__ esum, int N) {
  const int lane = threadIdx.x & 31;
  const int n = blockIdx.x * 8 + (threadIdx.x >> 5);
  if (n >= N) return;
  const float4 v = ((const float4*)xt)[(size_t)n * 32 + lane];
  const float4 as = ((const float4*)att_src)[lane];
  const float4 ad = ((const float4*)att_dst)[lane];
  float ps = v.x * as.x + v.y * as.y + v.z * as.z + v.w * as.w;
  float pd = v.x * ad.x + v.y * ad.y + v.z * ad.z + v.w * ad.w;
#pragma unroll
  for (int off = 1; off < 8; off <<= 1) {  // reduce within each head (8 lanes)
    ps += __shfl_xor(ps, off, 32);
    pd += __shfl_xor(pd, off, 32);
  }
  const int h = lane >> 3;
  if ((lane & 7) == 0) {
    a_src[(size_t)n * HEADS + h] = ps;
    a_dst[(size_t)n * HEADS + h] = pd;
  }
  if (lane < HEADS) {
    emax_u[(size_t)n * HEADS + lane] = 0u;  // below any real float's key
    esum[(size_t)n * HEADS + lane] = 0.f;
  }
  float4 zr = ((float4*)z)[(size_t)n * 32 + lane];
  const float4 gb = ((const float4*)gat_bias)[lane];
  const float4 rb = ((const float4*)res_b)[lane];
  zr.x += gb.x + rb.x; zr.y += gb.y + rb.y;
  zr.z += gb.z + rb.z; zr.w += gb.w + rb.w;
  ((float4*)z)[(size_t)n * 32 + lane] = zr;
}

static __device__ __forceinline__ void edge_sd(const int* __restrict__ ei, int e,
                                               int E, int& s, int& d) {
  if (e < E) { s = ei[e]; d = ei[E + e]; }
  else       { s = e - E; d = s; }  // self-loops appended
}

// K3: per-edge segment-max of leaky_relu(a_src[s]+a_dst[d]) per head
__global__ __launch_bounds__(256) void k_edge_max(const int* __restrict__ ei,
                                                  const float* __restrict__ a_src,
                                                  const float* __restrict__ a_dst,
                                                  unsigned* __restrict__ emax_u,
                                                  int E, int N) {
  const int e = blockIdx.x * blockDim.x + threadIdx.x;
  if (e >= E + N) return;
  int s, d; edge_sd(ei, e, E, s, d);
  const float4 as = ((const float4*)a_src)[s];
  const float4 ad = ((const float4*)a_dst)[d];
  unsigned* mp = emax_u + (size_t)d * HEADS;
  atomicMax(mp + 0, fkey(lrelu(as.x + ad.x)));
  atomicMax(mp + 1, fkey(lrelu(as.y + ad.y)));
  atomicMax(mp + 2, fkey(lrelu(as.z + ad.z)));
  atomicMax(mp + 3, fkey(lrelu(as.w + ad.w)));
}

// K4: ex = exp(e - emax[d]); stash per-edge ex; segment-sum into esum[d]
__global__ __launch_bounds__(256) void k_edge_sum(const int* __restrict__ ei,
                                                  const float* __restrict__ a_src,
                                                  const float* __restrict__ a_dst,
                                                  const unsigned* __restrict__ emax_u,
                                                  float* __restrict__ esum,
                                                  float* __restrict__ exb,
                                                  int E, int N) {
  const int e = blockIdx.x * blockDim.x + threadIdx.x;
  if (e >= E + N) return;
  int s, d; edge_sd(ei, e, E, s, d);
  const float4 as = ((const float4*)a_src)[s];
  const float4 ad = ((const float4*)a_dst)[d];
  const unsigned* mp = emax_u + (size_t)d * HEADS;
  float4 ex;
  ex.x = __expf(lrelu(as.x + ad.x) - fkey_inv(mp[0]));
  ex.y = __expf(lrelu(as.y + ad.y) - fkey_inv(mp[1]));
  ex.z = __expf(lrelu(as.z + ad.z) - fkey_inv(mp[2]));
  ex.w = __expf(lrelu(as.w + ad.w) - fkey_inv(mp[3]));
  ((float4*)exb)[e] = ex;
  float* sp = esum + (size_t)d * HEADS;
  atomicAdd(sp + 0, ex.x);
  atomicAdd(sp + 1, ex.y);
  atomicAdd(sp + 2, ex.z);
  atomicAdd(sp + 3, ex.w);
}

// K5: one wave per edge; 4 channels/lane; z[d] += alpha[h] * xt[s]
__global__ __launch_bounds__(256) void k_scatter(const int* __restrict__ ei,
                                                 const float* __restrict__ exb,
                                                 const float* __restrict__ esum,
                                                 const float* __restrict__ xt,
                                                 float* __restrict__ z,
                                                 int E, int N) {
  const int lane = threadIdx.x & 31;
  const int e = blockIdx.x * 8 + (threadIdx.x >> 5);
  if (e >= E + N) return;
  int s, d; edge_sd(ei, e, E, s, d);
  const int h = lane >> 3;
  const float alpha =
      exb[(size_t)e * HEADS + h] / (esum[(size_t)d * HEADS + h] + 1e-16f);
  const float4 m = ((const float4*)xt)[(size_t)s * 32 + lane];
  float* zp = z + (size_t)d * OUT_DIM + lane * 4;
  atomicAdd(zp + 0, alpha * m.x);
  atomicAdd(zp + 1, alpha * m.y);
  atomicAdd(zp + 2, alpha * m.z);
  atomicAdd(zp + 3, alpha * m.w);
}

// K6: LayerNorm over 128 + LeakyReLU; one wave per node
__global__ __launch_bounds__(256) void k_ln(const float* __restrict__ z,
                                            const float* __restrict__ ln_g,
                                            const float* __restrict__ ln_b,
                                            float* __restrict__ out, int N) {
  const int lane = threadIdx.x & 31;
  const int n = blockIdx.x * 8 + (threadIdx.x >> 5);
  if (n >= N) return;
  const float4 v = ((const float4*)z)[(size_t)n * 32 + lane];
  float s = v.x + v.y + v.z + v.w;
#pragma unroll
  for (int off = 1; off < 32; off <<= 1) s += __shfl_xor(s, off, 32);
  const float mu = s * (1.0f / 128.0f);
  const float dx = v.x - mu, dy = v.y - mu, dz = v.z - mu, dw = v.w - mu;
  float q = dx * dx + dy * dy + dz * dz + dw * dw;
#pragma unroll
  for (int off = 1; off < 32; off <<= 1) q += __shfl_xor(q, off, 32);
  const float inv = rsqrtf(q * (1.0f / 128.0f) + 1e-5f);
  const float4 g = ((const float4*)ln_g)[lane];
  const float4 b = ((const float4*)ln_b)[lane];
  float4 o;
  o.x = lrelu(dx * inv * g.x + b.x);
  o.y = lrelu(dy * inv * g.y + b.y);
  o.z = lrelu(dz * inv * g.z + b.z);
  o.w = lrelu(dw * inv * g.w + b.w);
  ((float4*)out)[(size_t)n * 32 + lane] = o;
}

extern "C" void kernel_launch(void* const* d_in, const int* in_sizes, int n_in,
                              void* d_out, int out_size, void* d_ws, size_t ws_size,
                              hipStream_t stream) {
  const float* x        = (const float*)d_in[0];
  const int*   ei       = (const int*)d_in[1];  // int32 (JAX x64 off), [2,E]
  const float* lin_w    = (const float*)d_in[2];
  const float* att_src  = (const float*)d_in[3];
  const float* att_dst  = (const float*)d_in[4];
  const float* gat_bias = (const float*)d_in[5];
  const float* res_w    = (const float*)d_in[6];
  const float* res_b    = (const float*)d_in[7];
  const float* ln_g     = (const float*)d_in[8];
  const float* ln_b     = (const float*)d_in[9];
  const int N  = in_sizes[0] / IN_CH;
  const int E  = in_sizes[1] / 2;
  const int Et = E + N;

  // workspace layout (floats): xt | z | a_src | a_dst | emax_u | esum | ex
  float*    xt     = (float*)d_ws;
  float*    z      = xt + (size_t)N * OUT_DIM;
  float*    a_src  = z + (size_t)N * OUT_DIM;
  float*    a_dst  = a_src + (size_t)N * HEADS;
  unsigned* emax_u = (unsigned*)(a_dst + (size_t)N * HEADS);
  float*    esum   = (float*)(emax_u + (size_t)N * HEADS);
  float*    exb    = esum + (size_t)N * HEADS;

  k_gemm<<<(N + 15) / 16, 256, 0, stream>>>(x, lin_w, res_w, xt, z, N);
  k_node<<<(N + 7) / 8, 256, 0, stream>>>(xt, att_src, att_dst, gat_bias, res_b,
                                          z, a_src, a_dst, emax_u, esum, N);
  k_edge_max<<<(Et + 255) / 256, 256, 0, stream>>>(ei, a_src, a_dst, emax_u, E, N);
  k_edge_sum<<<(Et + 255) / 256, 256, 0, stream>>>(ei, a_src, a_dst, emax_u, esum,
                                                   exb, E, N);
  k_scatter<<<(Et + 7) / 8, 256, 0, stream>>>(ei, exb, esum, xt, z, E, N);
  k_ln<<<(N + 7) / 8, 256, 0, stream>>>(z, ln_g, ln_b, (float*)d_out, N);
}